// Encoder_28424093565728
// MI455X (gfx1250) — compile-verified
//
#include <hip/hip_runtime.h>
#include <hip/hip_bf16.h>

// GCN 2-layer forward for MI455X (gfx1250, wave32).
// Dense projections use V_WMMA_F32_16X16X4_F32 (fp32 in/out, matches reference
// precision). Edge aggregation uses float atomics; h0/h1/out all fit in the
// 192MB L2 so gather+scatter traffic stays on-chip.

typedef __attribute__((ext_vector_type(2))) float v2f;
typedef __attribute__((ext_vector_type(8))) float v8f;

#define H0DIM 32
#define EMBDIM 128
#define FDIM 256

// ---------------- graph normalization ----------------

__global__ void k_init_deg(float* __restrict__ deg, int n) {
    int i = blockIdx.x * blockDim.x + threadIdx.x;
    if (i < n) deg[i] = 1.0f;   // self-loop weight
}

__global__ void k_deg_scatter(const int* __restrict__ row, const float* __restrict__ w,
                              float* __restrict__ deg, int E) {
    int e = blockIdx.x * blockDim.x + threadIdx.x;
    if (e < E) atomicAdd(&deg[row[e]], w[e]);
}

__global__ void k_dinv(float* __restrict__ deg, int n) {   // in-place deg -> dinv
    int i = blockIdx.x * blockDim.x + threadIdx.x;
    if (i < n) {
        float d = deg[i];
        deg[i] = (d > 0.0f) ? rsqrtf(fmaxf(d, 1e-12f)) : 0.0f;
    }
}

__global__ void k_normw(const int* __restrict__ row, const int* __restrict__ col,
                        const float* __restrict__ w, const float* __restrict__ dinv,
                        float* __restrict__ nw, int E) {
    int e = blockIdx.x * blockDim.x + threadIdx.x;
    if (e < E) nw[e] = dinv[row[e]] * w[e] * dinv[col[e]];
}

// ---------------- dense projection via fp32 WMMA ----------------
// D[nrows x NOUT] = A[nrows x K] @ B[K x NOUT], row-major, nrows % 16 == 0.
// One wave computes a 16-row tile across all NOUT columns.
// A-frag layout (16x4 f32): lane<16 holds K=0,1; lane>=16 holds K=2,3 (v[0],v[1]).
// B-frag mirrored; C/D: vgpr i -> row (i + 8*half), col = lane%16.

template <int K, int NOUT>
__global__ void k_gemm_wmma(const float* __restrict__ A, const float* __restrict__ B,
                            float* __restrict__ D, int nrows) {
    const int lane  = threadIdx.x & 31;
    const int wave  = threadIdx.x >> 5;
    const int mtile = blockIdx.x * (blockDim.x >> 5) + wave;
    if (mtile * 16 >= nrows) return;          // wave-uniform guard (EXEC stays all-1s)

    const int half = lane >> 4;
    const int mr   = lane & 15;
    const int m0   = mtile * 16;
    const float* arow = A + (size_t)(m0 + mr) * K;

#pragma unroll
    for (int nt = 0; nt < NOUT / 16; ++nt) {
        v8f acc = {};
#pragma unroll
        for (int kk = 0; kk < K / 4; ++kk) {
            const int kbase = kk * 4 + 2 * half;
            v2f a = *(const v2f*)(arow + kbase);          // 8B-aligned contiguous pair
            v2f b;
            b.x = B[(size_t)(kbase + 0) * NOUT + nt * 16 + mr];
            b.y = B[(size_t)(kbase + 1) * NOUT + nt * 16 + mr];
            acc = __builtin_amdgcn_wmma_f32_16x16x4_f32(
                false, a, false, b, (short)0, acc, false, false);
        }
#pragma unroll
        for (int i = 0; i < 8; ++i) {
            D[(size_t)(m0 + i + 8 * half) * NOUT + nt * 16 + mr] = acc[i];
        }
    }
}

// ---------------- aggregation ----------------

// Initialize accumulator with the self-loop term: agg = dinv[i]^2 * h[i].
template <int C>
__global__ void k_selfloop(const float* __restrict__ h, const float* __restrict__ dinv,
                           float* __restrict__ agg, int n) {
    int idx = blockIdx.x * blockDim.x + threadIdx.x;
    if (idx < n * C) {
        float dv = dinv[idx / C];
        agg[idx] = dv * dv * h[idx];
    }
}

// Per-edge scatter-add: C/4 lanes per edge, float4 gather + 4 float atomics.
template <int C>
__global__ void k_agg(const int* __restrict__ row, const int* __restrict__ col,
                      const float* __restrict__ nw, const float* __restrict__ h,
                      float* __restrict__ agg, int E) {
    const int LPE = C / 4;                          // lanes per edge
    int idx = blockIdx.x * blockDim.x + threadIdx.x;
    int e = idx / LPE;
    int l = idx % LPE;
    if (e >= E) return;
    float wv = nw[e];
    int c = col[e], r = row[e];
    const float4 hv = *(const float4*)(h + (size_t)c * C + l * 4);
    float* dst = agg + (size_t)r * C + l * 4;
    atomicAdd(dst + 0, wv * hv.x);
    atomicAdd(dst + 1, wv * hv.y);
    atomicAdd(dst + 2, wv * hv.z);
    atomicAdd(dst + 3, wv * hv.w);
}

template <int C>
__global__ void k_bias_relu(float* __restrict__ a, const float* __restrict__ b, int n) {
    int idx = blockIdx.x * blockDim.x + threadIdx.x;
    if (idx < n * C) {
        a[idx] = fmaxf(a[idx] + b[idx % C], 0.0f);
    }
}

// ---------------- launcher ----------------

extern "C" void kernel_launch(void* const* d_in, const int* in_sizes, int n_in,
                              void* d_out, int out_size, void* d_ws, size_t ws_size,
                              hipStream_t stream) {
    const float* x   = (const float*)d_in[0];
    const int*   ei  = (const int*)d_in[1];
    const float* ew  = (const float*)d_in[2];
    const float* W0  = (const float*)d_in[3];
    const float* b0  = (const float*)d_in[4];
    const float* W1  = (const float*)d_in[5];
    const float* b1  = (const float*)d_in[6];
    float* out = (float*)d_out;

    const int H0  = in_sizes[4];          // 32
    const int F   = in_sizes[3] / H0;     // 256
    const int N   = in_sizes[0] / F;      // 100000 (multiple of 16)
    const int E   = in_sizes[2];          // 3200000
    const int* row = ei;                  // edge_index[0]
    const int* col = ei + E;              // edge_index[1]

    // workspace layout (floats): dinv[N] | norm_w[E] | h0[N*32] | agg0[N*32] | h1[N*128]
    float* ws   = (float*)d_ws;
    float* dinv = ws;
    float* nw   = dinv + N;
    float* h0   = nw + E;
    float* agg0 = h0 + (size_t)N * H0DIM;
    float* h1   = agg0 + (size_t)N * H0DIM;

    const int TPB = 256;
    // --- normalization ---
    k_init_deg<<<(N + TPB - 1) / TPB, TPB, 0, stream>>>(dinv, N);
    k_deg_scatter<<<(E + TPB - 1) / TPB, TPB, 0, stream>>>(row, ew, dinv, E);
    k_dinv<<<(N + TPB - 1) / TPB, TPB, 0, stream>>>(dinv, N);
    k_normw<<<(E + TPB - 1) / TPB, TPB, 0, stream>>>(row, col, ew, dinv, nw, E);

    const int mtiles = (N + 15) / 16;                 // 6250
    const int gblk   = (mtiles + 3) / 4;              // 4 waves (128 thr) per block

    // --- layer 0 ---
    k_gemm_wmma<FDIM, H0DIM><<<gblk, 128, 0, stream>>>(x, W0, h0, N);
    k_selfloop<H0DIM><<<((size_t)N * H0DIM + TPB - 1) / TPB, TPB, 0, stream>>>(h0, dinv, agg0, N);
    k_agg<H0DIM><<<((size_t)E * (H0DIM / 4) + TPB - 1) / TPB, TPB, 0, stream>>>(row, col, nw, h0, agg0, E);
    k_bias_relu<H0DIM><<<((size_t)N * H0DIM + TPB - 1) / TPB, TPB, 0, stream>>>(agg0, b0, N);

    // --- layer 1 ---
    k_gemm_wmma<H0DIM, EMBDIM><<<gblk, 128, 0, stream>>>(agg0, W1, h1, N);
    k_selfloop<EMBDIM><<<((size_t)N * EMBDIM + TPB - 1) / TPB, TPB, 0, stream>>>(h1, dinv, out, N);
    k_agg<EMBDIM><<<((size_t)E * (EMBDIM / 4) + TPB - 1) / TPB, TPB, 0, stream>>>(row, col, nw, h1, out, E);
    k_bias_relu<EMBDIM><<<((size_t)N * EMBDIM + TPB - 1) / TPB, TPB, 0, stream>>>(out, b1, N);
}